// YoloLayer_24352464569088
// MI455X (gfx1250) — compile-verified
//
#include <hip/hip_runtime.h>
#include <hip/hip_bf16.h>
#include <stdint.h>

// ---- problem constants (match reference) ----
#define NBATCH 16
#define NANCH  3
#define NHH    76
#define NWW    76
#define HW     (NHH * NWW)          // 5776
#define NCLS   80
#define NCH    (5 + NCLS)           // 85
#define NTGT   50
#define REC    16                   // floats per target record in scratch/LDS
#define IGNORE_TH 0.5f

typedef __attribute__((ext_vector_type(2))) float v2f;
typedef __attribute__((ext_vector_type(8))) float v8f;

__device__ __forceinline__ float sigmoidf_(float x) {
    return 1.0f / (1.0f + expf(-x));
}

// safe_bce from reference: p in [0,1]
__device__ __forceinline__ float bce_(float p, float t) {
    float lp = fmaxf(logf(fmaxf(p, 1e-38f)), -100.0f);
    float ln = fmaxf(logf(fmaxf(1.0f - p, 1e-38f)), -100.0f);
    return -(t * lp + (1.0f - t) * ln);
}

// center-coordinate IoU, exact port of iou_cc
__device__ __forceinline__ float iou_cc_(float x1, float y1, float w1, float h1,
                                         float x2, float y2, float w2, float h2) {
    float wu = fmaxf(x1 + w1 * 0.5f, x2 + w2 * 0.5f) - fminf(x1 - w1 * 0.5f, x2 - w2 * 0.5f);
    float hu = fmaxf(y1 + h1 * 0.5f, y2 + h2 * 0.5f) - fminf(y1 - h1 * 0.5f, y2 - h2 * 0.5f);
    float wc = w1 + w2 - wu;
    float hc = h1 + h2 - hu;
    float inter = (wc <= 0.0f || hc <= 0.0f) ? 0.0f : wc * hc;
    float uni = w1 * h1 + w2 * h2 - inter;
    return inter / uni;
}

__device__ __forceinline__ float softplus_(float x) {
    // log(1 + e^x), stable
    return fmaxf(x, 0.0f) + log1pf(expf(-fabsf(x)));
}

__global__ void yolo_zero(float* out) {
    if (threadIdx.x == 0 && blockIdx.x == 0) out[0] = 0.0f;
}

// ---------------- kernel 1: per-target records ----------------
// record layout (16 floats):
// 0 valid | 1 gx | 2 gy | 3 gw | 4 gh | 5 best_n | 6 gi | 7 gj
// 8 tx | 9 ty | 10 tw | 11 th | 12 iou_gt | 13 cls_id | 14,15 pad
__global__ void yolo_targets(const float* __restrict__ out,
                             const float* __restrict__ tgt,
                             float* __restrict__ recs) {
    __shared__ float xflag[NTGT];
    const int b = blockIdx.x;
    const int t = threadIdx.x;

    float cls = 0.f, x = 0.f, y = 0.f, w = 0.f, h = 0.f;
    if (t < NTGT) {
        const float* p = tgt + (size_t)b * NTGT * 5 + (size_t)t * 5;
        cls = p[0]; x = p[1]; y = p[2]; w = p[3]; h = p[4];
        xflag[t] = (x != 0.0f) ? 1.0f : 0.0f;
    }
    __syncthreads();
    if (t >= NTGT) return;

    // prefix-AND (cumprod) validity
    float valid = 1.0f;
    for (int k = 0; k <= t; ++k) valid *= xflag[k];

    const float AW[3] = {10.0f, 16.0f, 33.0f};
    const float AH[3] = {13.0f, 30.0f, 23.0f};

    float gx = x * (float)NWW, gy = y * (float)NHH;
    float gw = w * 608.0f,     gh = h * 608.0f;

    // best anchor: argmax (first-index tie-break) of IoU of (0,0,aw,ah) vs (0,0,gw,gh)
    int bn = 0; float best = -1.0f;
    #pragma unroll
    for (int a = 0; a < 3; ++a) {
        float io = iou_cc_(0.f, 0.f, AW[a], AH[a], 0.f, 0.f, gw, gh);
        if (io > best) { best = io; bn = a; }
    }

    int gi = min(max((int)gx, 0), NWW - 1);
    int gj = min(max((int)gy, 0), NHH - 1);

    // pred box at the assigned cell
    size_t base = ((size_t)(b * NANCH + bn) * NCH) * HW + (size_t)gj * NWW + gi;
    float pxr = out[base];
    float pyr = out[base + (size_t)HW];
    float pwr = out[base + (size_t)2 * HW];
    float phr = out[base + (size_t)3 * HW];
    float px = sigmoidf_(pxr) + (float)gi;
    float py = sigmoidf_(pyr) + (float)gj;
    float pw = expf(pwr) * AW[bn];
    float ph = expf(phr) * AH[bn];
    float iou_gt = iou_cc_(gx, gy, gw, gh, px, py, pw, ph);

    float tw = (valid > 0.0f) ? logf(fmaxf(gw, 1e-12f) / AW[bn]) : 0.0f;
    float th = (valid > 0.0f) ? logf(fmaxf(gh, 1e-12f) / AH[bn]) : 0.0f;
    int cid = min(max((int)cls, 0), NCLS - 1);

    float* r = recs + (size_t)(b * NTGT + t) * REC;
    r[0] = valid; r[1] = gx; r[2] = gy; r[3] = gw; r[4] = gh;
    r[5] = (float)bn; r[6] = (float)gi; r[7] = (float)gj;
    r[8] = gx - (float)gi; r[9] = gy - (float)gj; r[10] = tw; r[11] = th;
    r[12] = iou_gt; r[13] = (float)cid; r[14] = 0.0f; r[15] = 0.0f;
}

// ---------------- kernel 2: dense streaming pass ----------------
// grid = (ceil(HW/256), NBATCH*NANCH), block = 256 (8 wave32)
__global__ void yolo_dense(const float* __restrict__ out,
                           const float* __restrict__ recs,
                           float* __restrict__ res) {
    __shared__ float srec[NTGT * REC];
    __shared__ float wsum[8];

    const int tid = threadIdx.x;
    const int ba  = blockIdx.y;
    const int b   = ba / NANCH;
    const int a   = ba - b * NANCH;

    // stage the 50 target records for this batch into LDS
    for (int idx = tid; idx < NTGT * REC; idx += blockDim.x)
        srec[idx] = recs[(size_t)b * NTGT * REC + idx];
    __syncthreads();

    const float AW[3] = {10.0f, 16.0f, 33.0f};
    const float AH[3] = {13.0f, 30.0f, 23.0f};

    const int cell = blockIdx.x * blockDim.x + tid;
    float local = 0.0f;

    if (cell < HW) {
        const int j = cell / NWW;
        const int i = cell - j * NWW;
        const size_t base = ((size_t)(b * NANCH + a) * NCH) * HW + cell;

        // only channels 0..4 are needed densely (5/85 of the tensor)
        float txr  = out[base];
        float tyr  = out[base + (size_t)HW];
        float twr  = out[base + (size_t)2 * HW];
        float thr  = out[base + (size_t)3 * HW];
        float cfr  = out[base + (size_t)4 * HW];

        float sx = sigmoidf_(txr);
        float sy = sigmoidf_(tyr);
        float px = sx + (float)i;
        float py = sy + (float)j;
        float pw = expf(twr) * AW[a];
        float ph = expf(thr) * AH[a];

        float maxiou = 0.0f;
        int last = -1;
        uint32_t cb0 = 0u, cb1 = 0u, cb2 = 0u;

        for (int t = 0; t < NTGT; ++t) {
            const float* r = &srec[t * REC];
            if (r[0] <= 0.5f) break;   // validity is a prefix mask (cumprod)
            float io = iou_cc_(px, py, pw, ph, r[1], r[2], r[3], r[4]);
            maxiou = fmaxf(maxiou, io);
            bool match = ((int)r[5] == a) && ((int)r[7] == j) && ((int)r[6] == i);
            if (match) {
                last = t;  // scan order: later valid target wins
                int c = (int)r[13];
                if      (c < 32) cb0 |= (1u << c);
                else if (c < 64) cb1 |= (1u << (c - 32));
                else             cb2 |= (1u << (c - 64));
            }
        }

        float conf_s = sigmoidf_(cfr);

        if (last >= 0) {
            const float* r = &srec[last * REC];
            // coord: BCE on sigmoid xy, MSE on raw wh
            local += bce_(sx, r[8]) + bce_(sy, r[9]);
            float dw = twr - r[10], dh = thr - r[11];
            local += dw * dw + dh * dh;
            // conf vs iou_gt
            local += bce_(conf_s, r[12]);
            // class loss: sparse 80-channel gather only at obj cells
            const float* clsp = out + base + (size_t)5 * HW;
            __builtin_prefetch(clsp, 0, 1);
            __builtin_prefetch(clsp + (size_t)40 * HW, 0, 1);
            for (int c = 0; c < NCLS; ++c) {
                float xc = clsp[(size_t)c * HW];
                uint32_t bit = (c < 32) ? (cb0 >> c)
                             : (c < 64) ? (cb1 >> (c - 32))
                                        : (cb2 >> (c - 64));
                local += softplus_(xc) - ((bit & 1u) ? xc : 0.0f);
            }
        } else if (maxiou <= IGNORE_TH) {
            // noobj cell: BCE(conf, 0)
            local += bce_(conf_s, 0.0f);
        }
        // cells with maxiou > thresh and no obj contribute 0 (mask == 0)
    }

    local *= (1.0f / (float)NBATCH);   // every loss term is divided by nB

    // ---- reduction: WMMA f32 16x16x4 with all-ones A folds 32 lanes -> 16 ----
    // Every lane puts its partial in B-VGPR0 (K-row r0), zero in B-VGPR1.
    // With A == ones, D[m][n] = colsum(n) = local(n) + local(n+16), so
    // D[0] in lane l holds local(l&15) + local((l&15)+16) for any K mapping.
    // EXEC is all ones here (no divergence at this point).
    v2f amat; amat.x = 1.0f; amat.y = 1.0f;
    v2f bmat; bmat.x = local; bmat.y = 0.0f;
    v8f cacc = {};
    cacc = __builtin_amdgcn_wmma_f32_16x16x4_f32(
        false, amat, false, bmat, (short)0, cacc, false, false);
    float wred = cacc[0];
    // finish: sum the 16 distinct column sums (xor within 16-lane halves)
    #pragma unroll
    for (int off = 1; off < 16; off <<= 1)
        wred += __shfl_xor(wred, off, 32);

    if ((tid & 31) == 0) wsum[tid >> 5] = wred;
    __syncthreads();
    if (tid == 0) {
        float s = 0.0f;
        #pragma unroll
        for (int w = 0; w < 8; ++w) s += wsum[w];
        atomicAdd(res, s);
    }
}

extern "C" void kernel_launch(void* const* d_in, const int* in_sizes, int n_in,
                              void* d_out, int out_size, void* d_ws, size_t ws_size,
                              hipStream_t stream) {
    const float* out = (const float*)d_in[0];   // (16, 255, 76, 76) f32
    const float* tgt = (const float*)d_in[1];   // (16, 250) f32
    float* res  = (float*)d_out;                // scalar loss
    float* recs = (float*)d_ws;                 // 16*50*16 floats = 51.2 KB

    yolo_zero<<<1, 1, 0, stream>>>(res);
    yolo_targets<<<NBATCH, 64, 0, stream>>>(out, tgt, recs);
    dim3 grid((HW + 255) / 256, NBATCH * NANCH);
    yolo_dense<<<grid, 256, 0, stream>>>(out, recs, res);
}